// Attn_90623809945974
// MI455X (gfx1250) — compile-verified
//
#include <hip/hip_runtime.h>
#include <math.h>

// Problem constants (from reference): H=512, B=64, L=4096, fp32 everywhere.
#define H 512
#define B 64
#define L 4096
#define LPW 16           // l-values handled per wave in the streaming kernel

typedef __attribute__((ext_vector_type(2))) float v2f;
typedef __attribute__((ext_vector_type(4))) float v4f;
typedef __attribute__((ext_vector_type(8))) float v8f;

// ---------------------------------------------------------------------------
// Kernel 1: u[b,h] = sum_o hidden[b,o] * W[o,h]   (M=64, N=512, K=512 GEMM)
// One wave32 per 16x16 output tile, K-loop of 128 x V_WMMA_F32_16X16X4_F32.
// A 16x4 layout: lanes 0-15 hold K={k0,k0+1} in {v.x,v.y}; lanes 16-31 hold
// K={k0+2,k0+3}. B 4x16 mirrored (lane = N). D: VGPR r -> row r+8*half.
// ---------------------------------------------------------------------------
__global__ __launch_bounds__(32)
void k_proj_wmma(const float* __restrict__ hidden,   // [B,H]
                 const float* __restrict__ Wm,       // [H,H] (o-major)
                 float* __restrict__ u) {            // [B,H]
    const int lane  = threadIdx.x;        // 0..31
    const int half  = lane >> 4;          // 0 -> K pair {0,1}, 1 -> {2,3}
    const int idx16 = lane & 15;          // M for A, N for B/D
    const int m0 = blockIdx.x * 16;       // b-tile
    const int n0 = blockIdx.y * 16;       // h-tile

    const float* __restrict__ arow = hidden + (size_t)(m0 + idx16) * H;

    v8f acc = {};
    #pragma unroll 4
    for (int k0 = 0; k0 < H; k0 += 4) {
        const int ka = k0 + 2 * half;
        v2f a, bm;
        a.x  = arow[ka];
        a.y  = arow[ka + 1];
        bm.x = Wm[(size_t)ka * H + n0 + idx16];
        bm.y = Wm[(size_t)(ka + 1) * H + n0 + idx16];
        acc = __builtin_amdgcn_wmma_f32_16x16x4_f32(
                  false, a, false, bm, (short)0, acc, false, false);
    }
    #pragma unroll
    for (int r = 0; r < 8; ++r) {
        u[(size_t)(m0 + r + 8 * half) * H + n0 + idx16] = acc[r];
    }
}

// ---------------------------------------------------------------------------
// Kernel 2: c[b] = hidden[b,:] . bias   (tiny)
// ---------------------------------------------------------------------------
__global__ __launch_bounds__(32)
void k_cbias(const float* __restrict__ hidden, const float* __restrict__ bias,
             float* __restrict__ c) {
    const int b = blockIdx.x;
    const int lane = threadIdx.x;
    const v4f* hb = (const v4f*)(hidden + (size_t)b * H);
    const v4f* bb = (const v4f*)bias;
    float acc = 0.f;
    #pragma unroll
    for (int j = 0; j < 4; ++j) {
        v4f h4 = hb[lane + j * 32];
        v4f b4 = bb[lane + j * 32];
        acc += h4.x * b4.x + h4.y * b4.y + h4.z * b4.z + h4.w * b4.w;
    }
    #pragma unroll
    for (int off = 16; off; off >>= 1) acc += __shfl_xor(acc, off, 32);
    if (lane == 0) c[b] = acc;
}

// ---------------------------------------------------------------------------
// Kernel 3 (the bandwidth-bound one): energies[b,l] = enc[l,b,:].u[b,:] + c[b]
// One wave per (b, 16 l's). u[b,:] hoisted to 16 regs/lane; enc streamed with
// non-temporal 128b loads (512 MB read exactly once, > L2 capacity).
// ---------------------------------------------------------------------------
__global__ __launch_bounds__(256)
void k_energy(const float* __restrict__ enc,   // [L,B,H]
              const float* __restrict__ u,     // [B,H]
              const float* __restrict__ c,     // [B]
              float* __restrict__ energies) {  // [B,L]
    const int lane = threadIdx.x & 31;
    const int wave = (int)((blockIdx.x * blockDim.x + threadIdx.x) >> 5);
    const int b  = wave / (L / LPW);
    const int l0 = (wave % (L / LPW)) * LPW;

    v4f ur[4];
    const v4f* ub = (const v4f*)(u + (size_t)b * H);
    #pragma unroll
    for (int j = 0; j < 4; ++j) ur[j] = ub[lane + j * 32];
    const float cb = c[b];

    for (int i = 0; i < LPW; ++i) {
        const int l = l0 + i;
        const v4f* e = (const v4f*)(enc + ((size_t)l * B + b) * H);
        float acc = 0.f;
        #pragma unroll
        for (int j = 0; j < 4; ++j) {
            v4f ev = __builtin_nontemporal_load(&e[lane + j * 32]);
            acc += ev.x * ur[j].x + ev.y * ur[j].y
                 + ev.z * ur[j].z + ev.w * ur[j].w;
        }
        #pragma unroll
        for (int off = 16; off; off >>= 1) acc += __shfl_xor(acc, off, 32);
        if (lane == 0) energies[(size_t)b * L + l] = acc + cb;
    }
}

// ---------------------------------------------------------------------------
// Kernel 4: softmax over L per b. 256 threads x 16 values in registers.
// ---------------------------------------------------------------------------
__global__ __launch_bounds__(256)
void k_softmax(const float* __restrict__ energies,  // [B,L]
               float* __restrict__ out) {           // [B,1,L]
    const int b = blockIdx.x;
    const int t = threadIdx.x;
    const int lane = t & 31;
    const int wid  = t >> 5;
    __shared__ float red[8];

    float v[16];
    float mx = -3.402823466e+38f;
    #pragma unroll
    for (int i = 0; i < 16; ++i) {
        v[i] = energies[(size_t)b * L + t + i * 256];
        mx = fmaxf(mx, v[i]);
    }
    #pragma unroll
    for (int off = 16; off; off >>= 1) mx = fmaxf(mx, __shfl_xor(mx, off, 32));
    if (lane == 0) red[wid] = mx;
    __syncthreads();
    mx = red[0];
    #pragma unroll
    for (int i = 1; i < 8; ++i) mx = fmaxf(mx, red[i]);
    __syncthreads();

    float s = 0.f;
    #pragma unroll
    for (int i = 0; i < 16; ++i) {
        v[i] = __expf(v[i] - mx);
        s += v[i];
    }
    #pragma unroll
    for (int off = 16; off; off >>= 1) s += __shfl_xor(s, off, 32);
    if (lane == 0) red[wid] = s;
    __syncthreads();
    s = red[0];
    #pragma unroll
    for (int i = 1; i < 8; ++i) s += red[i];
    const float inv = 1.0f / s;

    #pragma unroll
    for (int i = 0; i < 16; ++i) {
        out[(size_t)b * L + t + i * 256] = v[i] * inv;
    }
}

// ---------------------------------------------------------------------------
extern "C" void kernel_launch(void* const* d_in, const int* in_sizes, int n_in,
                              void* d_out, int out_size, void* d_ws, size_t ws_size,
                              hipStream_t stream) {
    const float* hidden = (const float*)d_in[0];  // [1,B,H]
    const float* enc    = (const float*)d_in[1];  // [L,B,H]
    const float* Wm     = (const float*)d_in[2];  // [H,H]
    const float* bias   = (const float*)d_in[3];  // [H]
    float* out = (float*)d_out;                   // [B,1,L]

    // Workspace layout (~1.13 MB total):
    //   u:        B*H floats  (128 KB)   @ 0
    //   c:        B   floats             @ 128 KB (256B-aligned slot)
    //   energies: B*L floats  (1 MB)     @ 128 KB + 256
    char* ws = (char*)d_ws;
    float* u        = (float*)ws;
    float* c        = (float*)(ws + (size_t)B * H * sizeof(float));
    float* energies = (float*)(ws + (size_t)B * H * sizeof(float) + 256);

    dim3 g1(B / 16, H / 16);                       // 4 x 32 tiles
    k_proj_wmma<<<g1, 32, 0, stream>>>(hidden, Wm, u);
    k_cbias<<<B, 32, 0, stream>>>(hidden, bias, c);

    const int n_waves  = B * (L / LPW);            // 16384 waves
    const int n_blocks = n_waves * 32 / 256;       // 2048 blocks of 8 waves
    k_energy<<<n_blocks, 256, 0, stream>>>(enc, u, c, energies);

    k_softmax<<<B, 256, 0, stream>>>(energies, out);
}